// LSTMParser_76836964926134
// MI455X (gfx1250) — compile-verified
//
#include <hip/hip_runtime.h>
#include <math.h>

typedef __attribute__((ext_vector_type(2))) float v2f;
typedef __attribute__((ext_vector_type(8))) float v8f;

#define LSTM_IN 125
#define LSTM_H  125
#define NG      500    // 4 * LSTM_H
#define SEQ     512
#define DH      250    // concat hidden dim
#define HSTRIDE 256    // padded row stride for h_concat
#define MLP_H   512
#define KP1     128    // padded K for input projection
#define KP2     256    // padded K for MLP projection

// ---------------------------------------------------------------------------
// Kernel 0: zero-pad operands so GEMM inner loops need no bounds checks.
//  Xp  (512x128)  <- X    (512x125)
//  Wfp (512x128)  <- Wih_f(500x125), rows 500..511 and cols 125..127 zeroed
//  Wbp (512x128)  <- Wih_b(500x125)
//  W1ap(512x256)  <- W1[:, :250],  cols 250..255 zeroed
//  W1bp(512x256)  <- W1[:, 250:],  cols 250..255 zeroed
// ---------------------------------------------------------------------------
__global__ void k_pack(const float* __restrict__ X,
                       const float* __restrict__ Wf,
                       const float* __restrict__ Wb,
                       const float* __restrict__ W1,
                       float* __restrict__ Xp,  float* __restrict__ Wfp,
                       float* __restrict__ Wbp, float* __restrict__ W1ap,
                       float* __restrict__ W1bp)
{
    const int idx = blockIdx.x * blockDim.x + threadIdx.x;   // 0 .. 512*256-1
    if (idx < SEQ * KP1) {
        const int r = idx >> 7, k = idx & (KP1 - 1);
        const bool kin = (k < LSTM_IN);
        Xp[idx]  = kin               ? X [r * LSTM_IN + k] : 0.0f;
        Wfp[idx] = (kin && r < NG)   ? Wf[r * LSTM_IN + k] : 0.0f;
        Wbp[idx] = (kin && r < NG)   ? Wb[r * LSTM_IN + k] : 0.0f;
    }
    if (idx < MLP_H * KP2) {
        const int r = idx >> 8, k = idx & (KP2 - 1);
        const bool kin = (k < DH);
        W1ap[idx] = kin ? W1[r * NG + k]      : 0.0f;
        W1bp[idx] = kin ? W1[r * NG + DH + k] : 0.0f;
    }
}

// ---------------------------------------------------------------------------
// Generic f32 WMMA GEMM:  C(MxN) = A(MxK) * B(NxK)^T, one wave per 16x16
// tile. Operands are pre-padded, so the K loop is branch-free: each step is
// one 8-byte A load + one 8-byte B load + one v_wmma_f32_16x16x4_f32.
// A-frag layout: M=lane&15, K=2*(lane>>4)+v.  C/D: M=v+8*(lane>>4), N=lane&15.
// ---------------------------------------------------------------------------
__global__ void k_wmma_gemm(const float* __restrict__ A,
                            const float* __restrict__ B,
                            float* __restrict__ C,
                            int lda, int ldb, int ldc, int K, int ncolLim)
{
    const int lane = threadIdx.x;                 // 0..31, wave32
    const int m0 = blockIdx.x * 16;
    const int n0 = blockIdx.y * 16;
    const int mr = m0 + (lane & 15);
    const int nc = n0 + (lane & 15);
    const int kh = (lane >> 4) * 2;               // K sub-block per half-wave

    const float* Ap = A + mr * lda + kh;          // 8-byte aligned (kh*4 in {0,8})
    const float* Bp = B + nc * ldb + kh;

    v8f acc = {};
#pragma unroll 8
    for (int kk = 0; kk < K; kk += 4) {
        const v2f a = *(const v2f*)(Ap + kk);
        const v2f b = *(const v2f*)(Bp + kk);
        acc = __builtin_amdgcn_wmma_f32_16x16x4_f32(false, a, false, b,
                                                    (short)0, acc, false, false);
    }
#pragma unroll
    for (int v = 0; v < 8; ++v) {
        const int row = m0 + v + 8 * (lane >> 4);
        const int col = n0 + (lane & 15);
        if (col < ncolLim) C[row * ldc + col] = acc[v];
    }
}

// ---------------------------------------------------------------------------
// Kernel 2: sequential bidirectional LSTM scan. 2 blocks (fwd/bwd), 512 thr.
// h kept in LDS; per-step: 500 threads each do a 125-wide dot (4 accumulators
// to break the serial FMA chain), then 125 threads apply gates and update c.
// ---------------------------------------------------------------------------
__global__ void k_lstm_scan(const float* __restrict__ Gf,
                            const float* __restrict__ Gb,
                            const float* __restrict__ Whh_f,
                            const float* __restrict__ Whh_b,
                            const float* __restrict__ bih_f,
                            const float* __restrict__ bhh_f,
                            const float* __restrict__ bih_b,
                            const float* __restrict__ bhh_b,
                            float* __restrict__ Hc)
{
    const int dir = blockIdx.x;                   // 0 = fwd, 1 = bwd
    const float* G   = dir ? Gb    : Gf;
    const float* Whh = dir ? Whh_b : Whh_f;
    const float* bih = dir ? bih_b : bih_f;
    const float* bhh = dir ? bhh_b : bhh_f;

    __shared__ float h_s[128];
    __shared__ float g_s[NG];

    const int tid = threadIdx.x;                  // 512 threads
    if (tid < 128) h_s[tid] = 0.0f;
    if (dir == 0 && tid < SEQ) {                  // zero padded cols 250..255 once
#pragma unroll
        for (int cIdx = DH; cIdx < HSTRIDE; ++cIdx) Hc[tid * HSTRIDE + cIdx] = 0.0f;
    }
    float cstate = 0.0f;
    const float bsum = (tid < NG) ? (bih[tid] + bhh[tid]) : 0.0f;
    __syncthreads();

    for (int s = 0; s < SEQ; ++s) {
        const int t = dir ? (SEQ - 1 - s) : s;    // time index into G / output
        if (tid < NG) {
            const float* wrow = Whh + tid * LSTM_H;
            float a0 = 0.f, a1 = 0.f, a2 = 0.f, a3 = 0.f;
            int k = 0;
            for (; k + 3 < LSTM_H; k += 4) {
                a0 = fmaf(wrow[k + 0], h_s[k + 0], a0);
                a1 = fmaf(wrow[k + 1], h_s[k + 1], a1);
                a2 = fmaf(wrow[k + 2], h_s[k + 2], a2);
                a3 = fmaf(wrow[k + 3], h_s[k + 3], a3);
            }
            for (; k < LSTM_H; ++k) a0 = fmaf(wrow[k], h_s[k], a0);
            g_s[tid] = G[t * NG + tid] + bsum + ((a0 + a1) + (a2 + a3));
        }
        __syncthreads();
        if (tid < LSTM_H) {
            const float gi = g_s[tid];
            const float gf = g_s[LSTM_H + tid];
            const float gc = g_s[2 * LSTM_H + tid];
            const float go = g_s[3 * LSTM_H + tid];
            const float ig = 1.0f / (1.0f + expf(-gi));
            const float fg = 1.0f / (1.0f + expf(-gf));
            const float gg = tanhf(gc);
            const float og = 1.0f / (1.0f + expf(-go));
            cstate = fg * cstate + ig * gg;
            const float h = og * tanhf(cstate);
            h_s[tid] = h;
            Hc[t * HSTRIDE + dir * LSTM_H + tid] = h;   // fwd: 0..124, bwd: 125..249
        }
        __syncthreads();
    }
}

// ---------------------------------------------------------------------------
// Kernel 4: fused biaffine reduction (never materializes the 512^3 tensor).
// out[i,j] = sum_k tanh(Pb[i,k] + Pa[j,k] + b1[k]) * W2[k] + b2.
// 16x16 output tile per 256-thread block; K chunked by 256 through LDS with
// row pad 260 (260 % 64 = 4) so the 16 j-lanes hit distinct LDS banks.
// Next chunk is prefetched (global_prefetch_b8) while tanh compute runs.
// ---------------------------------------------------------------------------
__global__ void k_biaffine(const float* __restrict__ Pa,
                           const float* __restrict__ Pb,
                           const float* __restrict__ b1,
                           const float* __restrict__ W2,
                           const float* __restrict__ b2,
                           float* __restrict__ out)
{
    const int KC = 256, PAD = 260;
    __shared__ float PbL[16 * 260];
    __shared__ float PaL[16 * 260];
    __shared__ float b1s[256];
    __shared__ float W2s[256];

    const int tid = threadIdx.x;                  // 256 threads
    const int i0 = blockIdx.y * 16;
    const int j0 = blockIdx.x * 16;
    const int il = tid >> 4;
    const int jl = tid & 15;

    float acc = 0.0f;
    for (int kc = 0; kc < MLP_H; kc += KC) {
        __syncthreads();                          // protect previous chunk's readers
        for (int idx = tid; idx < 16 * KC; idx += 256) {
            const int r = idx >> 8;               // KC == 256
            const int k = idx & 255;
            PbL[r * PAD + k] = Pb[(i0 + r) * MLP_H + kc + k];
            PaL[r * PAD + k] = Pa[(j0 + r) * MLP_H + kc + k];
        }
        if (tid < KC) { b1s[tid] = b1[kc + tid]; W2s[tid] = W2[kc + tid]; }
        __syncthreads();
        if (kc + KC < MLP_H && tid < 32) {        // prefetch next chunk rows
            const int r = tid & 15;
            if (tid < 16) __builtin_prefetch(&Pb[(i0 + r) * MLP_H + kc + KC], 0, 1);
            else          __builtin_prefetch(&Pa[(j0 + r) * MLP_H + kc + KC], 0, 1);
        }
        const float* pb = PbL + il * PAD;
        const float* pa = PaL + jl * PAD;
        for (int k = 0; k < KC; ++k)
            acc = fmaf(tanhf(pb[k] + pa[k] + b1s[k]), W2s[k], acc);
    }
    out[(i0 + il) * MLP_H + (j0 + jl)] = acc + b2[0];
}

// ---------------------------------------------------------------------------
extern "C" void kernel_launch(void* const* d_in, const int* in_sizes, int n_in,
                              void* d_out, int out_size, void* d_ws, size_t ws_size,
                              hipStream_t stream)
{
    (void)in_sizes; (void)n_in; (void)out_size; (void)ws_size;

    const float* x     = (const float*)d_in[0];   // (512,1,125)
    const float* Wih_f = (const float*)d_in[1];   // (500,125)
    const float* Whh_f = (const float*)d_in[2];   // (500,125)
    const float* bih_f = (const float*)d_in[3];
    const float* bhh_f = (const float*)d_in[4];
    const float* Wih_b = (const float*)d_in[5];
    const float* Whh_b = (const float*)d_in[6];
    const float* bih_b = (const float*)d_in[7];
    const float* bhh_b = (const float*)d_in[8];
    const float* W1    = (const float*)d_in[9];   // (512,500)
    const float* b1    = (const float*)d_in[10];  // (512,)
    const float* W2    = (const float*)d_in[11];  // (1,512)
    const float* b2    = (const float*)d_in[12];  // (1,)
    float* out = (float*)d_out;                   // (512,512)

    // Workspace layout (floats), ~6.2 MB total:
    float* ws   = (float*)d_ws;
    float* Xp   = ws;                      // 512*128
    float* Wfp  = Xp   + SEQ * KP1;        // 512*128
    float* Wbp  = Wfp  + SEQ * KP1;        // 512*128
    float* W1ap = Wbp  + SEQ * KP1;        // 512*256
    float* W1bp = W1ap + MLP_H * KP2;      // 512*256
    float* Gf   = W1bp + MLP_H * KP2;      // 512*500
    float* Gb   = Gf   + SEQ * NG;         // 512*500
    float* Hc   = Gb   + SEQ * NG;         // 512*256
    float* Pa   = Hc   + SEQ * HSTRIDE;    // 512*512
    float* Pb   = Pa   + SEQ * MLP_H;      // 512*512

    // 0) Pack / zero-pad GEMM operands (branch-free hot loops downstream).
    k_pack<<<(MLP_H * KP2 + 255) / 256, 256, 0, stream>>>(
        x, Wih_f, Wih_b, W1, Xp, Wfp, Wbp, W1ap, W1bp);

    // 1) Input projections: G = X @ Wih^T for both directions (WMMA f32).
    k_wmma_gemm<<<dim3(SEQ / 16, 32), 32, 0, stream>>>(Xp, Wfp, Gf,
                                                       KP1, KP1, NG, KP1, NG);
    k_wmma_gemm<<<dim3(SEQ / 16, 32), 32, 0, stream>>>(Xp, Wbp, Gb,
                                                       KP1, KP1, NG, KP1, NG);

    // 2) Sequential bidirectional scan (2 persistent blocks).
    k_lstm_scan<<<2, 512, 0, stream>>>(Gf, Gb, Whh_f, Whh_b,
                                       bih_f, bhh_f, bih_b, bhh_b, Hc);

    // 3) Pa / Pb projections via WMMA.
    k_wmma_gemm<<<dim3(SEQ / 16, MLP_H / 16), 32, 0, stream>>>(Hc, W1ap, Pa,
                                                               HSTRIDE, KP2, MLP_H, KP2, MLP_H);
    k_wmma_gemm<<<dim3(SEQ / 16, MLP_H / 16), 32, 0, stream>>>(Hc, W1bp, Pb,
                                                               HSTRIDE, KP2, MLP_H, KP2, MLP_H);

    // 4) Fused tanh-biaffine reduction.
    k_biaffine<<<dim3(MLP_H / 16, SEQ / 16), 256, 0, stream>>>(Pa, Pb, b1, W2, b2, out);
}